// EnhancedSyntaxGCN_31868657336589
// MI455X (gfx1250) — compile-verified
//
#include <hip/hip_runtime.h>

#define HID 64
#define NGRAPH 512
#define BN_EPS 1e-5f

typedef float v2f __attribute__((ext_vector_type(2)));
typedef float v8f __attribute__((ext_vector_type(8)));

// ---------------------------------------------------------------------------
// utility
// ---------------------------------------------------------------------------
__global__ void fill_f32(float* __restrict__ p, float v, long n) {
  long i = (long)blockIdx.x * blockDim.x + threadIdx.x;
  if (i < n) p[i] = v;
}

__global__ void deg_accum(const int* __restrict__ dst, float* __restrict__ deg, int E) {
  long e = (long)blockIdx.x * blockDim.x + threadIdx.x;
  if (e < E) atomicAdd(&deg[dst[e]], 1.0f);
}

__global__ void deg_rsqrt(float* __restrict__ deg, int N) {
  long i = (long)blockIdx.x * blockDim.x + threadIdx.x;
  if (i < N) deg[i] = rsqrtf(deg[i]);   // deg was initialized to 1 (self loop)
}

// ---------------------------------------------------------------------------
// layer-1 GEMM: hw = x(Nx3) @ W1(3x64)   (K=3: scalar, memory bound anyway)
// ---------------------------------------------------------------------------
__global__ void gemm_in3(const float* __restrict__ x, const float* __restrict__ W1,
                         float* __restrict__ hw, int N) {
  long t = (long)blockIdx.x * blockDim.x + threadIdx.x;
  long i = t >> 6;
  int c = (int)(t & 63);
  if (i >= N) return;
  float v = x[i * 3 + 0] * W1[0 * HID + c]
          + x[i * 3 + 1] * W1[1 * HID + c]
          + x[i * 3 + 2] * W1[2 * HID + c];
  hw[i * HID + c] = v;
}

// ---------------------------------------------------------------------------
// WMMA GEMM: hw = h(NPADx64) @ W(64x64), fp32, V_WMMA_F32_16X16X4_F32.
// One wave -> 16 rows x 64 cols (4 C-tiles); 8 waves/block -> 128 rows/block.
// A 16x4 frag : lane L holds A[m][k], m=L%16, k = vgpr + 2*(L/16)
// B 4x16 frag : lane L holds B[k][n], n=L%16, k = vgpr + 2*(L/16)
// C 16x16     : vgpr i, lane L -> row i + 8*(L/16), col L%16
//
// W is staged into LDS packed by k-pairs:
//   Wp[kpair][n] = { W[2*kpair][n], W[2*kpair+1][n] }   (8-byte elements)
// so each B fragment is a single aligned ds_load_b64 (no repacking movs).
// ---------------------------------------------------------------------------
__global__ __launch_bounds__(256) void gemm64_wmma(const float* __restrict__ h,
                                                   const float* __restrict__ W,
                                                   float* __restrict__ hw) {
  __shared__ v2f Wp[(HID / 2) * HID];          // 32 k-pairs x 64 cols, 16 KB
  const int tid = threadIdx.x;
  for (int i = tid; i < HID * HID; i += 256) {
    const int k = i >> 6;                      // 0..63
    const int n = i & 63;
    ((float*)Wp)[(k >> 1) * (2 * HID) + (n << 1) + (k & 1)] = W[i];
  }
  __syncthreads();

  const int wave = tid >> 5;
  const int lane = tid & 31;
  const int m    = lane & 15;
  const int hi   = lane >> 4;                  // 0 or 1
  const int g2   = hi << 1;                    // 0 or 2
  const long rowBase = (long)blockIdx.x * 128 + wave * 16;
  const float* __restrict__ arow = h + (rowBase + m) * HID;

  v8f c0 = {}, c1 = {}, c2 = {}, c3 = {};
#pragma unroll
  for (int k0 = 0; k0 < HID; k0 += 4) {
    v2f a;
    a.x = arow[k0 + g2 + 0];
    a.y = arow[k0 + g2 + 1];
    const int kp = (k0 >> 1) + hi;             // k-pair index for this lane half
    const v2f* __restrict__ bp = Wp + kp * HID + m;
    v2f b0 = bp[0];
    v2f b1 = bp[16];
    v2f b2 = bp[32];
    v2f b3 = bp[48];
    c0 = __builtin_amdgcn_wmma_f32_16x16x4_f32(false, a, false, b0, (short)0, c0, false, false);
    c1 = __builtin_amdgcn_wmma_f32_16x16x4_f32(false, a, false, b1, (short)0, c1, false, false);
    c2 = __builtin_amdgcn_wmma_f32_16x16x4_f32(false, a, false, b2, (short)0, c2, false, false);
    c3 = __builtin_amdgcn_wmma_f32_16x16x4_f32(false, a, false, b3, (short)0, c3, false, false);
  }

  const int rOff = hi * 8;
  const int col  = m;
  float* __restrict__ out = hw + rowBase * HID;
#pragma unroll
  for (int i = 0; i < 8; ++i) {
    const long r = rOff + i;
    out[r * HID + col +  0] = c0[i];
    out[r * HID + col + 16] = c1[i];
    out[r * HID + col + 32] = c2[i];
    out[r * HID + col + 48] = c3[i];
  }
}

// ---------------------------------------------------------------------------
// message scatter: agg[dst] += hw[src] * dinv[src]*dinv[dst]
// ---------------------------------------------------------------------------
__global__ void scatter_edges(const int* __restrict__ src, const int* __restrict__ dst,
                              const float* __restrict__ dinv, const float* __restrict__ hw,
                              float* __restrict__ agg, int E) {
  long t = (long)blockIdx.x * blockDim.x + threadIdx.x;
  long e = t >> 6;
  int c = (int)(t & 63);
  if (e >= E) return;
  int s = src[e], d = dst[e];
  float w = dinv[s] * dinv[d];
  atomicAdd(&agg[(long)d * HID + c], hw[(long)s * HID + c] * w);
}

// self-loop term + bias: agg += hw * dinv^2 + b
__global__ void self_bias(const float* __restrict__ hw, const float* __restrict__ dinv,
                          const float* __restrict__ b, float* __restrict__ agg, int N) {
  long t = (long)blockIdx.x * blockDim.x + threadIdx.x;
  long i = t >> 6;
  int c = (int)(t & 63);
  if (i >= N) return;
  float di = dinv[i];
  agg[i * HID + c] += hw[i * HID + c] * di * di + b[c];
}

// ---------------------------------------------------------------------------
// batchnorm (64 channels over N rows): partial sums per block -> atomics
// ---------------------------------------------------------------------------
__global__ __launch_bounds__(256) void bn_stats(const float* __restrict__ h,
                                                float* __restrict__ sum,
                                                float* __restrict__ sumsq, int N) {
  __shared__ float ss[256], sq[256];
  const int c = threadIdx.x & 63;
  const int g = threadIdx.x >> 6;          // 0..3
  const int row0 = blockIdx.x * 256;
  int rowEnd = row0 + 256; if (rowEnd > N) rowEnd = N;
  float s = 0.f, q = 0.f;
  for (int i = row0 + g; i < rowEnd; i += 4) {
    float v = h[(long)i * HID + c];
    s += v; q += v * v;
  }
  ss[threadIdx.x] = s; sq[threadIdx.x] = q;
  __syncthreads();
  if (threadIdx.x < 64) {
    s = ss[c] + ss[64 + c] + ss[128 + c] + ss[192 + c];
    q = sq[c] + sq[64 + c] + sq[128 + c] + sq[192 + c];
    atomicAdd(&sum[c], s);
    atomicAdd(&sumsq[c], q);
  }
}

__global__ void bn_finalize(const float* __restrict__ sum, const float* __restrict__ sumsq,
                            const float* __restrict__ gamma, const float* __restrict__ beta,
                            float* __restrict__ scale, float* __restrict__ shift, float invN) {
  int c = threadIdx.x;
  float mu  = sum[c] * invN;
  float var = sumsq[c] * invN - mu * mu;
  float sc  = gamma[c] * rsqrtf(var + BN_EPS);
  scale[c] = sc;
  shift[c] = beta[c] - mu * sc;
}

// apply BN + ReLU; also zero pad rows [N, NPAD) so the WMMA GEMM reads zeros
__global__ void bn_apply(const float* __restrict__ pre, const float* __restrict__ scale,
                         const float* __restrict__ shift, float* __restrict__ out,
                         int N, int NPAD) {
  long t = (long)blockIdx.x * blockDim.x + threadIdx.x;
  long i = t >> 6;
  int c = (int)(t & 63);
  if (i >= NPAD) return;
  float v = 0.f;
  if (i < N) {
    v = pre[i * HID + c] * scale[c] + shift[c];
    v = fmaxf(v, 0.f);
  }
  out[i * HID + c] = v;
}

// ---------------------------------------------------------------------------
// global mean/max pooling over sorted batch ids
// ---------------------------------------------------------------------------
__global__ void pool_count(const int* __restrict__ batch, float* __restrict__ counts, int N) {
  long i = (long)blockIdx.x * blockDim.x + threadIdx.x;
  if (i < N) atomicAdd(&counts[batch[i]], 1.0f);
}

__global__ void pool_accum(const float* __restrict__ h, const int* __restrict__ batch,
                           float* __restrict__ mean, unsigned* __restrict__ maxb, int N) {
  long t = (long)blockIdx.x * blockDim.x + threadIdx.x;
  long i = t >> 6;
  int c = (int)(t & 63);
  if (i >= N) return;
  int g = batch[i];
  float v = h[i * HID + c];                    // v >= 0 (post-ReLU)
  atomicAdd(&mean[(long)g * HID + c], v);
  atomicMax(&maxb[(long)g * HID + c], __float_as_uint(v));  // valid for v >= 0
}

__global__ void pool_finish(const float* __restrict__ counts, const float* __restrict__ mean,
                            const unsigned* __restrict__ maxb, float* __restrict__ gfeat) {
  long t = (long)blockIdx.x * blockDim.x + threadIdx.x;   // NGRAPH*64 threads
  if (t >= (long)NGRAPH * HID) return;
  long g = t >> 6;
  int c = (int)(t & 63);
  float cnt = fmaxf(counts[g], 1.0f);
  gfeat[g * (2 * HID) + c]       = mean[g * HID + c] / cnt;
  gfeat[g * (2 * HID) + HID + c] = __uint_as_float(maxb[g * HID + c]);
}

// ---------------------------------------------------------------------------
// head MLP (tiny: 512 rows)
// ---------------------------------------------------------------------------
__global__ void head_gemm(const float* __restrict__ A, const float* __restrict__ W,
                          const float* __restrict__ bias, float* __restrict__ out,
                          int K, int F) {
  long t = (long)blockIdx.x * blockDim.x + threadIdx.x;
  long g = t / F;
  int c = (int)(t % F);
  if (g >= NGRAPH) return;
  float acc = bias[c];
  for (int k = 0; k < K; ++k) acc += A[g * K + k] * W[k * F + c];
  out[g * F + c] = acc;
}

// one block of F threads: per-channel BN over 512 rows + ReLU
__global__ void head_bn(const float* __restrict__ pre, const float* __restrict__ gamma,
                        const float* __restrict__ beta, float* __restrict__ out, int F) {
  int c = threadIdx.x;
  float s = 0.f, q = 0.f;
  for (int g = 0; g < NGRAPH; ++g) {
    float v = pre[g * F + c];
    s += v; q += v * v;
  }
  const float inv = 1.0f / (float)NGRAPH;
  float mu  = s * inv;
  float var = q * inv - mu * mu;
  float sc  = gamma[c] * rsqrtf(var + BN_EPS);
  float sh  = beta[c] - mu * sc;
  for (int g = 0; g < NGRAPH; ++g)
    out[g * F + c] = fmaxf(pre[g * F + c] * sc + sh, 0.f);
}

__global__ void final_lin(const float* __restrict__ a, const float* __restrict__ W,
                          const float* __restrict__ b, float* __restrict__ out) {
  long t = (long)blockIdx.x * blockDim.x + threadIdx.x;  // NGRAPH*2 threads
  if (t >= (long)NGRAPH * 2) return;
  long g = t >> 1;
  int j = (int)(t & 1);
  float acc = b[j];
  for (int k = 0; k < 32; ++k) acc += a[g * 32 + k] * W[k * 2 + j];
  out[g * 2 + j] = acc;
}

// ---------------------------------------------------------------------------
static inline unsigned gridFor(long n, int b) { return (unsigned)((n + b - 1) / b); }

extern "C" void kernel_launch(void* const* d_in, const int* in_sizes, int n_in,
                              void* d_out, int out_size, void* d_ws, size_t ws_size,
                              hipStream_t stream) {
  const int N = in_sizes[2];            // batch has N entries
  const int E = in_sizes[1] / 2;        // edge_index is [2, E]
  const int NPAD = (N + 127) & ~127;

  const float* x    = (const float*)d_in[0];
  const int*   src  = (const int*)d_in[1];
  const int*   dstp = src + E;
  const int*   batch = (const int*)d_in[2];
  const float* W1 = (const float*)d_in[3];
  const float* b1 = (const float*)d_in[4];
  const float* W2 = (const float*)d_in[5];
  const float* b2 = (const float*)d_in[6];
  const float* W3 = (const float*)d_in[7];
  const float* b3 = (const float*)d_in[8];
  const float* bng[3] = {(const float*)d_in[9],  (const float*)d_in[11], (const float*)d_in[13]};
  const float* bnb[3] = {(const float*)d_in[10], (const float*)d_in[12], (const float*)d_in[14]};
  const float* lin1W = (const float*)d_in[15];
  const float* lin1b = (const float*)d_in[16];
  const float* lin2W = (const float*)d_in[17];
  const float* lin2b = (const float*)d_in[18];
  const float* lin3W = (const float*)d_in[19];
  const float* lin3b = (const float*)d_in[20];
  const float* bnf1g = (const float*)d_in[21];
  const float* bnf1b = (const float*)d_in[22];
  const float* bnf2g = (const float*)d_in[23];
  const float* bnf2b = (const float*)d_in[24];

  // workspace layout (floats)
  float* ws = (float*)d_ws;
  size_t off = 0;
  float* dinv   = ws + off; off += (size_t)NPAD;
  float* h0     = ws + off; off += (size_t)NPAD * HID;
  float* h1     = ws + off; off += (size_t)NPAD * HID;
  float* hw     = ws + off; off += (size_t)NPAD * HID;
  float* bnSum  = ws + off; off += 64;
  float* bnSq   = ws + off; off += 64;
  float* bnScale= ws + off; off += 64;
  float* bnShift= ws + off; off += 64;
  float* counts = ws + off; off += NGRAPH;
  float* pmean  = ws + off; off += (size_t)NGRAPH * HID;
  float* pmax   = ws + off; off += (size_t)NGRAPH * HID;
  float* gfeat  = ws + off; off += (size_t)NGRAPH * 2 * HID;
  float* gtmp   = ws + off; off += (size_t)NGRAPH * HID;
  float* g1     = ws + off; off += (size_t)NGRAPH * HID;
  float* g2buf  = ws + off; off += (size_t)NGRAPH * 32;
  (void)ws_size; (void)n_in; (void)out_size;

  const long NC = (long)N * HID;

  // degrees -> dinv = rsqrt(deg + 1)
  fill_f32<<<gridFor(N, 256), 256, 0, stream>>>(dinv, 1.0f, N);
  deg_accum<<<gridFor(E, 256), 256, 0, stream>>>(dstp, dinv, E);
  deg_rsqrt<<<gridFor(N, 256), 256, 0, stream>>>(dinv, N);

  const float* Wmat[3] = {W1, W2, W3};
  const float* bvec[3] = {b1, b2, b3};
  for (int l = 0; l < 3; ++l) {
    // hw = h @ W
    if (l == 0)
      gemm_in3<<<gridFor(NC, 256), 256, 0, stream>>>(x, W1, hw, N);
    else
      gemm64_wmma<<<(unsigned)(NPAD / 128), 256, 0, stream>>>(h0, Wmat[l], hw);
    // zero aggregation buffer + BN stats
    fill_f32<<<gridFor(NC, 256), 256, 0, stream>>>(h1, 0.0f, NC);
    fill_f32<<<1, 128, 0, stream>>>(bnSum, 0.0f, 128);   // bnSum + bnSq contiguous
    // agg = scatter(msg) + self + bias
    scatter_edges<<<gridFor((long)E * HID, 256), 256, 0, stream>>>(src, dstp, dinv, hw, h1, E);
    self_bias<<<gridFor(NC, 256), 256, 0, stream>>>(hw, dinv, bvec[l], h1, N);
    // batchnorm + relu -> h0 (pad rows zeroed)
    bn_stats<<<gridFor(N, 256), 256, 0, stream>>>(h1, bnSum, bnSq, N);
    bn_finalize<<<1, 64, 0, stream>>>(bnSum, bnSq, bng[l], bnb[l], bnScale, bnShift, 1.0f / (float)N);
    bn_apply<<<gridFor((long)NPAD * HID, 256), 256, 0, stream>>>(h1, bnScale, bnShift, h0, N, NPAD);
  }

  // pooling
  fill_f32<<<gridFor(NGRAPH, 256), 256, 0, stream>>>(counts, 0.0f, NGRAPH);
  fill_f32<<<gridFor((long)NGRAPH * HID * 2, 256), 256, 0, stream>>>(pmean, 0.0f, (long)NGRAPH * HID * 2); // pmean+pmax contiguous
  pool_count<<<gridFor(N, 256), 256, 0, stream>>>(batch, counts, N);
  pool_accum<<<gridFor(NC, 256), 256, 0, stream>>>(h0, batch, pmean, (unsigned*)pmax, N);
  pool_finish<<<gridFor((long)NGRAPH * HID, 256), 256, 0, stream>>>(counts, pmean, (const unsigned*)pmax, gfeat);

  // head MLP
  head_gemm<<<gridFor((long)NGRAPH * 64, 256), 256, 0, stream>>>(gfeat, lin1W, lin1b, gtmp, 128, 64);
  head_bn<<<1, 64, 0, stream>>>(gtmp, bnf1g, bnf1b, g1, 64);
  head_gemm<<<gridFor((long)NGRAPH * 32, 256), 256, 0, stream>>>(g1, lin2W, lin2b, gtmp, 64, 32);
  head_bn<<<1, 32, 0, stream>>>(gtmp, bnf2g, bnf2b, g2buf, 32);
  final_lin<<<gridFor((long)NGRAPH * 2, 256), 256, 0, stream>>>(g2buf, lin3W, lin3b, (float*)d_out);
}